// SpatialModulation_79637283602999
// MI455X (gfx1250) — compile-verified
//
#include <hip/hip_runtime.h>
#include <math.h>

typedef __attribute__((ext_vector_type(2))) float v2f;
typedef __attribute__((ext_vector_type(4))) float v4f;
typedef __attribute__((ext_vector_type(8))) float v8f;

#define NB   16        // batch
#define CCH  128       // channels
#define HH   256
#define WW   256
#define HWSZ 65536     // H*W (2^16)

// ---------------------------------------------------------------------------
// Pass 1: per-pixel channel sum (via WMMA ones-matvec) and channel max.
// One wave32 owns 16 consecutive pixels. A-tile (16 pixels x 4 channels) is
// loaded in the exact V_WMMA_F32_16X16X4_F32 A-layout:
//   lane l (m16 = l&15, half = l>>4): VGPR0 = x[c0+2*half][pix], VGPR1 = x[c0+2*half+1][pix]
// B = all-ones 4x16 -> D[m,n] = sum_c x[c,m] for every n.
// ---------------------------------------------------------------------------
__global__ void __launch_bounds__(256)
sm_reduce_kernel(const float* __restrict__ x,
                 float* __restrict__ amean,
                 float* __restrict__ mmax) {
  const int wave = (int)((blockIdx.x * blockDim.x + threadIdx.x) >> 5);
  const int lane = (int)(threadIdx.x & 31u);
  const int b    = wave >> 12;            // 4096 waves per batch image
  const int q0   = (wave & 4095) << 4;    // first of 16 pixels
  const int m16  = lane & 15;
  const int half = lane >> 4;

  const float* xb = x + (size_t)b * CCH * HWSZ + (size_t)q0 + m16;

  v8f acc = {};
  v2f onesv; onesv[0] = 1.0f; onesv[1] = 1.0f;
  float mx = -__builtin_inff();

#pragma unroll 4
  for (int c0 = 0; c0 < CCH; c0 += 4) {
    v2f a;
    a[0] = __builtin_nontemporal_load(xb + (size_t)(c0 + 2 * half)     * HWSZ);
    a[1] = __builtin_nontemporal_load(xb + (size_t)(c0 + 2 * half + 1) * HWSZ);
    mx = fmaxf(mx, fmaxf(a[0], a[1]));
    // D = A(16x4) * ones(4x16) + C : accumulates channel sums on the matrix pipe
    acc = __builtin_amdgcn_wmma_f32_16x16x4_f32(
        /*neg_a=*/false, a, /*neg_b=*/false, onesv,
        /*c_mod=*/(short)0, acc, /*reuse_a=*/false, /*reuse_b=*/false);
  }

  // combine the two half-wave partial maxima (wave32: lanes l <-> l^16)
  mx = fmaxf(mx, __shfl_xor(mx, 16, 32));

  // D layout: VGPR r, lanes 0-15 -> row r; lanes 16-31 -> row r+8.
  // Lane l (with (l&15) < 8) owns row (l&7) + 8*half in element (l&7).
  const int r = lane & 7;
  float sum = acc[0];
  sum = (r == 1) ? acc[1] : sum;
  sum = (r == 2) ? acc[2] : sum;
  sum = (r == 3) ? acc[3] : sum;
  sum = (r == 4) ? acc[4] : sum;
  sum = (r == 5) ? acc[5] : sum;
  sum = (r == 6) ? acc[6] : sum;
  sum = (r == 7) ? acc[7] : sum;

  if (m16 < 8)
    amean[((size_t)b << 16) + q0 + r + 8 * half] = sum * (1.0f / 128.0f);
  if (lane < 16)
    mmax[((size_t)b << 16) + q0 + lane] = mx;
}

// ---------------------------------------------------------------------------
// Pass 2: 4-neighbor stencil over the (mean, max) planes + sigmoid gate.
// Tiny (1M pixels); planes are L2-resident.
// ---------------------------------------------------------------------------
__global__ void __launch_bounds__(256)
sm_stencil_kernel(const float* __restrict__ amean,
                  const float* __restrict__ mmax,
                  const float* __restrict__ wc,
                  float* __restrict__ sgate) {
  const int idx = (int)(blockIdx.x * blockDim.x + threadIdx.x); // B*HW
  const int b = idx >> 16;
  const int q = idx & (HWSZ - 1);
  const int h = q >> 8;
  const int wv = q & 255;

  float g = 0.0f;
#pragma unroll
  for (int k = 0; k < 2; ++k) {
    const float* y = (k == 0 ? amean : mmax) + ((size_t)b << 16);
    const float p0 = wc[b * 8 + 4 * k + 0];
    const float p1 = wc[b * 8 + 4 * k + 1];
    const float p2 = wc[b * 8 + 4 * k + 2];
    const float p3 = wc[b * 8 + 4 * k + 3];
    const float up    = (h  > 0)       ? y[q - WW] : 0.0f;
    const float down  = (h  < HH - 1)  ? y[q + WW] : 0.0f;
    const float left  = (wv > 0)       ? y[q - 1]  : 0.0f;
    const float right = (wv < WW - 1)  ? y[q + 1]  : 0.0f;
    const float ctr   = y[q];
    g += p0 * up + p1 * down + p2 * left + p3 * right
       - (p0 + p1 + p2 + p3) * ctr;
  }
  sgate[idx] = 1.0f / (1.0f + expf(-g));
}

// ---------------------------------------------------------------------------
// Pass 3: out = x * sigmoid(g), b128 NT loads/stores for x/out, cached loads
// for the 4 MiB gate plane (128x reuse from L2).
// ---------------------------------------------------------------------------
__global__ void __launch_bounds__(256)
sm_modulate_kernel(const float* __restrict__ x,
                   const float* __restrict__ sgate,
                   float* __restrict__ out) {
  const size_t i = ((size_t)blockIdx.x * blockDim.x + threadIdx.x) << 2;
  const int b = (int)(i >> 23);          // C*HW = 2^23
  const int q = (int)(i & (HWSZ - 1));   // multiple of 4, same row for all 4

  v4f xv = __builtin_nontemporal_load((const v4f*)(x + i));
  v4f sv = *(const v4f*)(sgate + ((size_t)b << 16) + q);
  v4f ov = xv * sv;
  __builtin_nontemporal_store(ov, (v4f*)(out + i));
}

// ---------------------------------------------------------------------------
extern "C" void kernel_launch(void* const* d_in, const int* in_sizes, int n_in,
                              void* d_out, int out_size, void* d_ws, size_t ws_size,
                              hipStream_t stream) {
  const float* x  = (const float*)d_in[0];   // (16,128,256,256) f32
  const float* wc = (const float*)d_in[1];   // (16,8) f32
  float* out = (float*)d_out;

  float* amean = (float*)d_ws;               // 1,048,576 floats (4 MiB)
  float* mmax  = amean + (size_t)NB * HWSZ;  // 4 MiB
  float* sgate = mmax  + (size_t)NB * HWSZ;  // 4 MiB

  // Pass 1: 65536 waves (one per 16-pixel tile) = 2,097,152 threads
  sm_reduce_kernel<<<8192, 256, 0, stream>>>(x, amean, mmax);

  // Pass 2: 1,048,576 pixels
  sm_stencil_kernel<<<4096, 256, 0, stream>>>(amean, mmax, wc, sgate);

  // Pass 3: 134,217,728 / 4 elements per thread
  sm_modulate_kernel<<<131072, 256, 0, stream>>>(x, sgate, out);
}